// GNN_node_59433757442359
// MI455X (gfx1250) — compile-verified
//
#include <hip/hip_runtime.h>
#include <hip/hip_bf16.h>

#define N_NODES 80000
#define N_EDGES 1280000
#define IN_DIM  32
#define EDGE_DIM 16
#define HID     64
#define HID2    128
#define DEPTH   4

typedef float v2f __attribute__((ext_vector_type(2)));
typedef float v8f __attribute__((ext_vector_type(8)));

// D = A(16x4 f32) * B(4x16 f32) + C, wave32, one wave per 16x16 tile.
__device__ __forceinline__ v8f wmma4(v2f a, v2f b, v8f c) {
    return __builtin_amdgcn_wmma_f32_16x16x4_f32(
        /*neg_a=*/false, a, /*neg_b=*/false, b,
        /*c_mod=*/(short)0, c, /*reuse_a=*/false, /*reuse_b=*/false);
}

// ---------------------------------------------------------------------------
// h = relu(x @ Wx + bx)   [80000,32]@[32,64] -> d_out (h)
// block = 128 threads (4 waves), each wave owns a 16-col tile; 16 rows/block.
__global__ void k_node_embed(const float* __restrict__ x,
                             const float* __restrict__ Wx,
                             const float* __restrict__ bx,
                             float* __restrict__ h) {
    __shared__ float xs[16 * 36];   // stride 36: 144B rows (16B aligned), banks 36m%64 distinct
    const int tid = threadIdx.x;
    const int mb  = blockIdx.x * 16;
    {   // 16x32 tile = 128 float4, one per thread
        int m = tid >> 3, kc = (tid & 7) * 4;
        *(float4*)&xs[m * 36 + kc] =
            *(const float4*)&x[(size_t)(mb + m) * IN_DIM + kc];
    }
    __syncthreads();
    const int lane = tid & 31, wave = tid >> 5;
    const int half = lane >> 4, l16 = lane & 15;
    const int nb = wave * 16;
    v8f acc = {};
    for (int ks = 0; ks < IN_DIM / 4; ++ks) {
        const int k0 = ks * 4 + 2 * half;
        v2f a, b;
        a[0] = xs[l16 * 36 + k0];
        a[1] = xs[l16 * 36 + k0 + 1];
        b[0] = Wx[(size_t)k0 * HID + nb + l16];
        b[1] = Wx[(size_t)(k0 + 1) * HID + nb + l16];
        acc = wmma4(a, b, acc);
    }
    const int col = nb + l16;
    const float bias = bx[col];
#pragma unroll
    for (int v = 0; v < 8; ++v) {
        const int row = mb + v + 8 * half;
        float z = acc[v] + bias;
        h[(size_t)row * HID + col] = z > 0.f ? z : 0.f;
    }
}

// ---------------------------------------------------------------------------
__global__ void k_zero4(float4* __restrict__ p, int n4) {
    int i = blockIdx.x * blockDim.x + threadIdx.x;
    if (i < n4) p[i] = make_float4(0.f, 0.f, 0.f, 0.f);
}

// ---------------------------------------------------------------------------
// Fused: ea = relu(edge_attr@We+be); msg = relu(h[row]+ea); atomic scatter-add
// into agg[col]. 16 threads per edge, 4 output channels per thread.
__global__ void k_message(const float* __restrict__ h,
                          const int* __restrict__ ei,
                          const float* __restrict__ edge_attr,
                          const float* __restrict__ We,
                          const float* __restrict__ be,
                          float* __restrict__ agg) {
    __shared__ float Ws[EDGE_DIM * HID];
    __shared__ float bs[HID];
    const int tid = threadIdx.x;  // 256
    {   // 16x64 = 256 float4, one per thread
        *(float4*)&Ws[tid * 4] = *(const float4*)&We[tid * 4];
    }
    if (tid < HID) bs[tid] = be[tid];
    __syncthreads();

    const int gt = blockIdx.x * 256 + tid;
    const int e  = gt >> 4;
    const int g  = (gt & 15) * 4;  // channel base: 0,4,...,60
    if (e >= N_EDGES) return;

    const int row = ei[e];
    const int col = ei[N_EDGES + e];

    // edge features: 16 floats = 4 x float4 (wave-uniform per 16-lane group,
    // coalesces/broadcasts from L0)
    float ev[EDGE_DIM];
    {
        const float4* ef4 = (const float4*)(edge_attr + (size_t)e * EDGE_DIM);
#pragma unroll
        for (int q = 0; q < 4; ++q) {
            float4 t = ef4[q];
            ev[q * 4 + 0] = t.x; ev[q * 4 + 1] = t.y;
            ev[q * 4 + 2] = t.z; ev[q * 4 + 3] = t.w;
        }
    }
    const float4 hv = *(const float4*)(h + (size_t)row * HID + g);
    const float  hr[4] = {hv.x, hv.y, hv.z, hv.w};
    float* ag = agg + (size_t)col * HID + g;
#pragma unroll
    for (int j = 0; j < 4; ++j) {
        float acc = bs[g + j];
#pragma unroll
        for (int k = 0; k < EDGE_DIM; ++k)
            acc = fmaf(ev[k], Ws[k * HID + g + j], acc);
        acc = acc > 0.f ? acc : 0.f;    // relu(edge embedding)
        float m = hr[j] + acc;
        m = m > 0.f ? m : 0.f;          // relu(message)
        atomicAdd(&ag[j], m);
    }
}

// ---------------------------------------------------------------------------
// z1 = ((1+eps)*h + agg) @ W1 + b1  -> z1[N,128]; per-column sum/sumsq atomics.
// block = 256 threads (8 waves), one 16x128 row-tile per block.
__global__ void k_gemm1(const float* __restrict__ h,
                        const float* __restrict__ agg,
                        const float* __restrict__ W1l,
                        const float* __restrict__ b1l,
                        const float* __restrict__ epsp, int layer,
                        float* __restrict__ z1,
                        float* __restrict__ sum, float* __restrict__ sumsq) {
    __shared__ float us[16 * 68];   // stride 68: 272B rows (16B aligned), banks 4m%64 distinct
    const int tid = threadIdx.x;  // 256
    const int mb  = blockIdx.x * 16;
    const float epl = 1.0f + epsp[layer];
    {   // 16x64 tile = 256 float4, one per thread
        int m = tid >> 4, kc = (tid & 15) * 4;
        size_t idx = (size_t)(mb + m) * HID + kc;
        float4 hv = *(const float4*)&h[idx];
        float4 av = *(const float4*)&agg[idx];
        float4 u = make_float4(fmaf(epl, hv.x, av.x), fmaf(epl, hv.y, av.y),
                               fmaf(epl, hv.z, av.z), fmaf(epl, hv.w, av.w));
        *(float4*)&us[m * 68 + kc] = u;
    }
    __syncthreads();
    const int lane = tid & 31, wave = tid >> 5;  // 8 waves -> 8 n-tiles (N=128)
    const int half = lane >> 4, l16 = lane & 15;
    const int nb = wave * 16;
    v8f acc = {};
    for (int ks = 0; ks < HID / 4; ++ks) {
        const int k0 = ks * 4 + 2 * half;
        v2f a, b;
        a[0] = us[l16 * 68 + k0];
        a[1] = us[l16 * 68 + k0 + 1];
        b[0] = W1l[(size_t)k0 * HID2 + nb + l16];
        b[1] = W1l[(size_t)(k0 + 1) * HID2 + nb + l16];
        acc = wmma4(a, b, acc);
    }
    const int col = nb + l16;
    const float bias = b1l[col];
    float s = 0.f, q = 0.f;
#pragma unroll
    for (int v = 0; v < 8; ++v) {
        float z = acc[v] + bias;
        z1[(size_t)(mb + v + 8 * half) * HID2 + col] = z;
        s += z;
        q += z * z;
    }
    // lanes L and L+16 hold the same column (different rows): fold halves.
    s += __shfl_down(s, 16, 32);
    q += __shfl_down(q, 16, 32);
    if (half == 0) {
        atomicAdd(&sum[col], s);
        atomicAdd(&sumsq[col], q);
    }
}

// ---------------------------------------------------------------------------
// Fold BN stats + gamma/beta into per-column scale/shift.
__global__ void k_bnfin(const float* __restrict__ sum,
                        const float* __restrict__ sumsq,
                        const float* __restrict__ g1l,
                        const float* __restrict__ bt1l,
                        float* __restrict__ scale, float* __restrict__ shift) {
    const int c = threadIdx.x;  // 128
    const float inv_n = 1.0f / (float)N_NODES;
    float mu  = sum[c] * inv_n;
    float var = sumsq[c] * inv_n - mu * mu;
    float sc  = g1l[c] * rsqrtf(var + 1e-5f);
    scale[c] = sc;
    shift[c] = bt1l[c] - mu * sc;
}

// ---------------------------------------------------------------------------
// h = LN( relu(BN(z1)) @ W2 + b2 ) [+relu]; block = 128 threads (4 waves).
__global__ void k_gemm2(const float* __restrict__ z1,
                        const float* __restrict__ scale,
                        const float* __restrict__ shift,
                        const float* __restrict__ W2l,
                        const float* __restrict__ b2l,
                        const float* __restrict__ lng,
                        const float* __restrict__ lnb,
                        int relu_out, float* __restrict__ h) {
    __shared__ float ys[16 * 132];  // stride 132: 528B rows (16B aligned), banks 4m%64 distinct
    __shared__ float os[16 * 68];
    __shared__ float lm[16], lr[16];
    const int tid = threadIdx.x;  // 128
    const int mb  = blockIdx.x * 16;
    {   // 16x128 tile = 512 float4, 4 per thread; BN + ReLU applied on the fly
        for (int i = tid; i < 16 * HID2 / 4; i += 128) {
            int m = i >> 5, cc = (i & 31) * 4;
            float4 zv = *(const float4*)&z1[(size_t)(mb + m) * HID2 + cc];
            float4 sc = *(const float4*)&scale[cc];
            float4 sh = *(const float4*)&shift[cc];
            float4 y;
            y.x = fmaf(zv.x, sc.x, sh.x); y.y = fmaf(zv.y, sc.y, sh.y);
            y.z = fmaf(zv.z, sc.z, sh.z); y.w = fmaf(zv.w, sc.w, sh.w);
            y.x = y.x > 0.f ? y.x : 0.f;  y.y = y.y > 0.f ? y.y : 0.f;
            y.z = y.z > 0.f ? y.z : 0.f;  y.w = y.w > 0.f ? y.w : 0.f;
            *(float4*)&ys[m * 132 + cc] = y;
        }
    }
    __syncthreads();
    const int lane = tid & 31, wave = tid >> 5;  // 4 waves -> 4 n-tiles (N=64)
    const int half = lane >> 4, l16 = lane & 15;
    const int nb = wave * 16;
    v8f acc = {};
    for (int ks = 0; ks < HID2 / 4; ++ks) {
        const int k0 = ks * 4 + 2 * half;
        v2f a, b;
        a[0] = ys[l16 * 132 + k0];
        a[1] = ys[l16 * 132 + k0 + 1];
        b[0] = W2l[(size_t)k0 * HID + nb + l16];
        b[1] = W2l[(size_t)(k0 + 1) * HID + nb + l16];
        acc = wmma4(a, b, acc);
    }
    const int col = nb + l16;
    const float bias = b2l[col];
#pragma unroll
    for (int v = 0; v < 8; ++v)
        os[(v + 8 * half) * 68 + col] = acc[v] + bias;
    __syncthreads();
    if (tid < 16) {  // per-row LayerNorm stats over 64 cols
        float s = 0.f;
        for (int c = 0; c < HID; ++c) s += os[tid * 68 + c];
        float mean = s * (1.0f / HID);
        float q = 0.f;
        for (int c = 0; c < HID; ++c) {
            float d = os[tid * 68 + c] - mean;
            q += d * d;
        }
        lm[tid] = mean;
        lr[tid] = rsqrtf(q * (1.0f / HID) + 1e-5f);
    }
    __syncthreads();
    // normalize + optional relu, write float4 (16x64 = 256 float4, 2/thread)
    for (int i = tid; i < 16 * HID / 4; i += 128) {
        int m = i >> 4, cc = (i & 15) * 4;
        float mean = lm[m], rstd = lr[m];
        float4 ov = *(const float4*)&os[m * 68 + cc];
        float4 gv = *(const float4*)&lng[cc];
        float4 bv = *(const float4*)&lnb[cc];
        float4 r;
        r.x = (ov.x - mean) * rstd * gv.x + bv.x;
        r.y = (ov.y - mean) * rstd * gv.y + bv.y;
        r.z = (ov.z - mean) * rstd * gv.z + bv.z;
        r.w = (ov.w - mean) * rstd * gv.w + bv.w;
        if (relu_out) {
            r.x = r.x > 0.f ? r.x : 0.f;  r.y = r.y > 0.f ? r.y : 0.f;
            r.z = r.z > 0.f ? r.z : 0.f;  r.w = r.w > 0.f ? r.w : 0.f;
        }
        *(float4*)&h[(size_t)(mb + m) * HID + cc] = r;
    }
}

// ---------------------------------------------------------------------------
extern "C" void kernel_launch(void* const* d_in, const int* in_sizes, int n_in,
                              void* d_out, int out_size, void* d_ws, size_t ws_size,
                              hipStream_t stream) {
    const float* x   = (const float*)d_in[0];
    const int*   ei  = (const int*)d_in[1];
    const float* ea  = (const float*)d_in[2];
    const float* Wx  = (const float*)d_in[3];
    const float* bx  = (const float*)d_in[4];
    const float* We  = (const float*)d_in[5];
    const float* be  = (const float*)d_in[6];
    const float* W1  = (const float*)d_in[7];
    const float* b1  = (const float*)d_in[8];
    const float* g1  = (const float*)d_in[9];
    const float* bt1 = (const float*)d_in[10];
    const float* W2  = (const float*)d_in[11];
    const float* b2  = (const float*)d_in[12];
    const float* eps = (const float*)d_in[13];
    const float* lng = (const float*)d_in[14];
    const float* lnb = (const float*)d_in[15];

    float* h = (float*)d_out;  // h lives in d_out, updated in place each layer

    // workspace layout: agg | stats(512) | z1
    float* agg   = (float*)d_ws;
    float* stats = agg + (size_t)N_NODES * HID;
    float* sum   = stats;
    float* sumsq = stats + 128;
    float* scale = stats + 256;
    float* shift = stats + 384;
    float* z1    = stats + 512;

    k_node_embed<<<N_NODES / 16, 128, 0, stream>>>(x, Wx, bx, h);

    for (int l = 0; l < DEPTH; ++l) {
        const int n4 = (N_NODES * HID + 512) / 4;  // agg + stats are contiguous
        k_zero4<<<(n4 + 255) / 256, 256, 0, stream>>>((float4*)agg, n4);
        k_message<<<(N_EDGES * 16) / 256, 256, 0, stream>>>(h, ei, ea, We, be, agg);
        k_gemm1<<<N_NODES / 16, 256, 0, stream>>>(
            h, agg, W1 + (size_t)l * HID * HID2, b1 + (size_t)l * HID2,
            eps, l, z1, sum, sumsq);
        k_bnfin<<<1, 128, 0, stream>>>(
            sum, sumsq, g1 + (size_t)l * HID2, bt1 + (size_t)l * HID2, scale, shift);
        k_gemm2<<<N_NODES / 16, 128, 0, stream>>>(
            z1, scale, shift, W2 + (size_t)l * HID2 * HID, b2 + (size_t)l * HID,
            lng + (size_t)l * HID, lnb + (size_t)l * HID,
            (l < DEPTH - 1) ? 1 : 0, h);
    }
}